// GroupPearson_24429773979799
// MI455X (gfx1250) — compile-verified
//
#include <hip/hip_runtime.h>

// Grouped Pearson correlation, bandwidth-bound segmented reduction.
// CDNA5 strategy: all 4096 groups x 6 f32 stats = 96 KB fits in the 320 KB
// WGP LDS -> full LDS privatization with no-return ds_add_f32 atomics.
// SoA layout s[k*4096 + g]: 4096 % 64 == 0 so each atomic's bank is g%64,
// spreading a wave's 32 random lanes across all 64 LDS banks (the AoS g*6
// layout hit only even banks). Single global_atomic_add_f32 flush per
// block, non-temporal b128 streaming loads + global_prefetch_b8.

typedef float v4f __attribute__((ext_vector_type(4)));
typedef int   v4i __attribute__((ext_vector_type(4)));

#define NGROUPS 4096
#define NSTATS  6
#define NACC    (NGROUPS * NSTATS)   // 24576 floats = 96 KB

__device__ __forceinline__ void lds_fadd(float* p, float v) {
  // relaxed, workgroup scope -> plain ds_add_f32 (no return, no fence)
  __hip_atomic_fetch_add(p, v, __ATOMIC_RELAXED, __HIP_MEMORY_SCOPE_WORKGROUP);
}
__device__ __forceinline__ void glob_fadd(float* p, float v) {
  // relaxed, agent scope -> global_atomic_add_f32 (no return, no fence)
  __hip_atomic_fetch_add(p, v, __ATOMIC_RELAXED, __HIP_MEMORY_SCOPE_AGENT);
}

// SoA: stat k for group g lives at s[k*NGROUPS + g]
__device__ __forceinline__ void accum_elem(float* s, int gi, float xv, float y) {
  lds_fadd(s + 0 * NGROUPS + gi, 1.0f);      // n
  lds_fadd(s + 1 * NGROUPS + gi, xv);        // sum x  (exp)
  lds_fadd(s + 2 * NGROUPS + gi, y);         // sum y  (pred)
  lds_fadd(s + 3 * NGROUPS + gi, xv * y);    // sum xy
  lds_fadd(s + 4 * NGROUPS + gi, xv * xv);   // sum xx
  lds_fadd(s + 5 * NGROUPS + gi, y * y);     // sum yy
}

// First function in the file -> visible in the disasm snippet.
__global__ __launch_bounds__(256) void gp_accum(const float* __restrict__ pred,
                                                const float* __restrict__ expv,
                                                const int*   __restrict__ group,
                                                float* __restrict__ ws,
                                                int n) {
  __shared__ float s[NACC];

  // compact zero-init: 24 contiguous ds_store_b128 per thread
  {
    v4f* s4 = (v4f*)s;
    int base = (int)threadIdx.x * (NACC / 4 / 256);   // 24 v4 slots each
#pragma unroll
    for (int k = 0; k < NACC / 4 / 256; ++k) {
      v4f z = {0.0f, 0.0f, 0.0f, 0.0f};
      s4[base + k] = z;
    }
  }
  __syncthreads();

  const int n8 = n >> 3;            // each iteration handles 8 elements
  const v4f* __restrict__ p4 = (const v4f*)pred;
  const v4f* __restrict__ x4 = (const v4f*)expv;
  const v4i* __restrict__ g4 = (const v4i*)group;

  int tid    = blockIdx.x * 256 + threadIdx.x;
  int stride = gridDim.x * 256;

  for (int i = tid; i < n8; i += stride) {
    int base = i << 1;
    int pf   = (i + stride) << 1;   // prefetch next grid-stride chunk
    __builtin_prefetch((const void*)(p4 + pf), 0, 0);   // global_prefetch_b8
    __builtin_prefetch((const void*)(x4 + pf), 0, 0);
    __builtin_prefetch((const void*)(g4 + pf), 0, 0);

    // streamed once, bigger than L2 -> non-temporal b128 loads
    v4f p0 = __builtin_nontemporal_load(p4 + base);
    v4f p1 = __builtin_nontemporal_load(p4 + base + 1);
    v4f x0 = __builtin_nontemporal_load(x4 + base);
    v4f x1 = __builtin_nontemporal_load(x4 + base + 1);
    v4i g0 = __builtin_nontemporal_load(g4 + base);
    v4i g1 = __builtin_nontemporal_load(g4 + base + 1);

#pragma unroll
    for (int j = 0; j < 4; ++j) accum_elem(s, g0[j], x0[j], p0[j]);
#pragma unroll
    for (int j = 0; j < 4; ++j) accum_elem(s, g1[j], x1[j], p1[j]);
  }

  // tail (n not a multiple of 8): block 0 handles it into its LDS copy
  if (blockIdx.x == 0) {
    int rem = n & 7;
    if ((int)threadIdx.x < rem) {
      int idx = (n8 << 3) + (int)threadIdx.x;
      accum_elem(s, group[idx], expv[idx], pred[idx]);
    }
  }

  __syncthreads();
  // single flush of the privatized accumulators to global scratch (same SoA)
  for (int i = threadIdx.x; i < NACC; i += 256)
    glob_fadd(&ws[i], s[i]);
}

__global__ __launch_bounds__(256) void gp_zero(float* __restrict__ ws) {
  int i = blockIdx.x * 256 + threadIdx.x;
  if (i < NACC) ws[i] = 0.0f;
}

__global__ __launch_bounds__(256) void gp_final(const float* __restrict__ ws,
                                                float* __restrict__ out,
                                                float inv_total) {
  __shared__ float red[256];
  float acc = 0.0f;
  for (int g = threadIdx.x; g < NGROUPS; g += 256) {
    float nn  = ws[0 * NGROUPS + g];
    float sx  = ws[1 * NGROUPS + g];
    float sy  = ws[2 * NGROUPS + g];
    float sxy = ws[3 * NGROUPS + g];
    float sxx = ws[4 * NGROUPS + g];
    float syy = ws[5 * NGROUPS + g];
    float ns  = nn > 0.0f ? nn : 1.0f;
    float mx  = sx / ns;
    float my  = sy / ns;
    float cov = sxy / ns - mx * my;
    float vx  = sxx / ns - mx * mx;
    float vy  = syy / ns - my * my;
    float den = sqrtf(fmaxf(vx * vy, 0.0f));
    float corr = den > 0.0f ? (cov / den) : 0.0f;
    acc += corr * nn;   // size-weighted
  }
  red[threadIdx.x] = acc;
  __syncthreads();
  for (int off = 128; off > 0; off >>= 1) {
    if ((int)threadIdx.x < off) red[threadIdx.x] += red[threadIdx.x + off];
    __syncthreads();
  }
  if (threadIdx.x == 0) out[0] = -red[0] * inv_total;  // sum(n) == N exactly
}

extern "C" void kernel_launch(void* const* d_in, const int* in_sizes, int n_in,
                              void* d_out, int out_size, void* d_ws, size_t ws_size,
                              hipStream_t stream) {
  const float* pred  = (const float*)d_in[0];
  const float* expv  = (const float*)d_in[1];
  const int*   group = (const int*)d_in[2];   // integer input -> const int* per contract
  int n = in_sizes[0];
  float* ws = (float*)d_ws;   // 96 KB of group stats (SoA)

  gp_zero <<<(NACC + 255) / 256, 256, 0, stream>>>(ws);
  gp_accum<<<512, 256, 0, stream>>>(pred, expv, group, ws, n);
  gp_final<<<1, 256, 0, stream>>>(ws, (float*)d_out, 1.0f / (float)n);
}